// Hash1SH_67568425501374
// MI455X (gfx1250) — compile-verified
//
#include <hip/hip_runtime.h>
#include <math.h>

// ---------------- problem constants ----------------
#define NLVL 16          // L
#define TSIZE (1u << 19) // hash table entries per level
#define FSTRIDE 36       // LDS feature-row stride (floats): 4*(9m mod 16) -> conflict-free
#define WSTRIDE 36       // LDS weight-row stride
#define NWAVES 4
#define BLOCK (NWAVES * 32)

typedef __attribute__((ext_vector_type(2))) float v2f;
typedef __attribute__((ext_vector_type(8))) float v8f;

// RES[l] = floor(16 * 2^(l/3))  (SCALE = 32^(1/15) = 2^(1/3))
__constant__ float c_res[NLVL] = {16.f, 20.f, 25.f, 32.f, 40.f, 50.f, 64.f, 80.f,
                                  101.f, 128.f, 161.f, 203.f, 256.f, 322.f, 406.f, 512.f};

__device__ __forceinline__ v8f wmma_f32_k4(v2f a, v2f b, v8f c) {
  // D = A(16x4) * B(4x16) + C(16x16), full fp32 — matches reference precision.
  return __builtin_amdgcn_wmma_f32_16x16x4_f32(false, a, false, b, (short)0, c, false, false);
}

// One dense layer for the wave's 32-sample tile pair, in-place in LDS.
// buf: [32][FSTRIDE] activations (rows = samples). wT: [32][WSTRIDE], wT[n][k] = W[k][n].
template <int NT, bool RELU>
__device__ __forceinline__ void mlp_layer(float* __restrict__ buf,
                                          const float* __restrict__ wT,
                                          const float* __restrict__ bias, int lane) {
  const int l16 = lane & 15;
  const int half = lane >> 4; // lanes 0-15 hold K+{0,1}, lanes 16-31 hold K+{2,3}
#pragma unroll
  for (int tile = 0; tile < 2; ++tile) {
    // Pull the entire 16x32 A tile into registers before overwriting its rows.
    const float* arow = buf + (tile * 16 + l16) * FSTRIDE + half * 2;
    v2f a[8];
#pragma unroll
    for (int kk = 0; kk < 8; ++kk) a[kk] = *(const v2f*)(arow + kk * 4);

#pragma unroll
    for (int nt = 0; nt < NT; ++nt) {
      const int col = nt * 16 + l16;
      const float* wrow = wT + col * WSTRIDE + half * 2;
      v8f acc = {0.f, 0.f, 0.f, 0.f, 0.f, 0.f, 0.f, 0.f};
#pragma unroll
      for (int kk = 0; kk < 8; ++kk) {
        v2f b = *(const v2f*)(wrow + kk * 4);
        acc = wmma_f32_k4(a[kk], b, acc);
      }
      const float bv = bias[col];
#pragma unroll
      for (int j = 0; j < 8; ++j) { // C/D layout: VGPR j -> row j + 8*half, col = l16
        float v = acc[j] + bv;
        if (RELU) v = fmaxf(v, 0.0f);
        buf[(tile * 16 + j + half * 8) * FSTRIDE + col] = v;
      }
    }
  }
}

__global__ __launch_bounds__(BLOCK) void Hash1SH_fused_kernel(
    const float* __restrict__ xs, const float* __restrict__ ds,
    const float* __restrict__ embx, const float* __restrict__ embw,
    const float* __restrict__ lw1, const float* __restrict__ lb1,
    const float* __restrict__ lw2, const float* __restrict__ lb2,
    const float* __restrict__ lw3, const float* __restrict__ lb3,
    const float* __restrict__ ww1, const float* __restrict__ wb1,
    const float* __restrict__ ww2, const float* __restrict__ wb2,
    const float* __restrict__ ww3, const float* __restrict__ wb3,
    float* __restrict__ out, int n) {
  __shared__ float sW[6][32][WSTRIDE]; // transposed, zero-padded weights
  __shared__ float sB[6][32];          // zero-padded biases
  __shared__ float sF[NWAVES][2][32][FSTRIDE]; // per-wave activation tiles (X-net, W-net)

  const int tid = threadIdx.x;
  const int wave = tid >> 5;
  const int lane = tid & 31;

  // ---- stage weights (transposed) + biases into LDS ----
  {
    const float* gw[6] = {lw1, lw2, lw3, ww1, ww2, ww3};
    const float* gb[6] = {lb1, lb2, lb3, wb1, wb2, wb3};
    const int od[6] = {32, 32, 16, 32, 32, 30};
#pragma unroll
    for (int m = 0; m < 6; ++m) {
      for (int idx = tid; idx < 32 * WSTRIDE; idx += BLOCK) {
        const int nn = idx / WSTRIDE;
        const int kk = idx % WSTRIDE;
        float v = 0.f;
        if (kk < 32 && nn < od[m]) v = gw[m][kk * od[m] + nn];
        sW[m][nn][kk] = v;
      }
      if (tid < 32) sB[m][tid] = (tid < od[m]) ? gb[m][tid] : 0.f;
    }
  }
  __syncthreads();

  const int s = blockIdx.x * BLOCK + tid;
  const int sc = (s < n) ? s : (n - 1); // clamp so WMMA tiles stay full/EXEC all-1s

  // ---- phase 1: hash-grid encode (shared indices for both tables) ----
  // reference: xs_ = 2*xs; xin = (xs_+1)/2 = xs + 0.5
  const float x0 = xs[3 * sc + 0] + 0.5f;
  const float x1 = xs[3 * sc + 1] + 0.5f;
  const float x2 = xs[3 * sc + 2] + 0.5f;

  float* fX = &sF[wave][0][lane][0];
  float* fW = &sF[wave][1][lane][0];
  const float2* ex2 = (const float2*)embx;
  const float2* ew2 = (const float2*)embw;

  for (int l = 0; l < NLVL; ++l) {
    const float rs = c_res[l];
    const float p0 = x0 * rs, p1 = x1 * rs, p2 = x2 * rs;
    const float f0 = floorf(p0), f1 = floorf(p1), f2 = floorf(p2);
    const float r0 = p0 - f0, r1 = p1 - f1, r2 = p2 - f2;
    const unsigned u0 = (unsigned)f0, u1 = (unsigned)f1, u2 = (unsigned)f2;
    const float w0[2] = {1.f - r0, r0};
    const float w1a[2] = {1.f - r1, r1};
    const float w2a[2] = {1.f - r2, r2};
    const float2* tx = ex2 + ((size_t)l << 19);
    const float2* tw = ew2 + ((size_t)l << 19);
    float ax = 0.f, ay = 0.f, bx = 0.f, by = 0.f;
#pragma unroll
    for (int c = 0; c < 8; ++c) {
      const unsigned o0 = (c >> 2) & 1u, o1 = (c >> 1) & 1u, o2 = c & 1u;
      const unsigned h = (u0 + o0) * 1u ^ (u1 + o1) * 2654435761u ^ (u2 + o2) * 805459861u;
      const unsigned idx = h & (TSIZE - 1u);
      const float w = w0[o0] * w1a[o1] * w2a[o2];
      const float2 vx = tx[idx]; // 8B gathers, L2-resident (tables = 128MB < 192MB L2)
      const float2 vw = tw[idx];
      ax += w * vx.x; ay += w * vx.y;
      bx += w * vw.x; by += w * vw.y;
    }
    fX[2 * l + 0] = ax; fX[2 * l + 1] = ay;
    fW[2 * l + 0] = bx; fW[2 * l + 1] = by;
  }

  // ---- phase 2: two tiny MLPs via fp32 WMMA (in-place in LDS) ----
  float* bufX = &sF[wave][0][0][0];
  float* bufW = &sF[wave][1][0][0];
  mlp_layer<2, true >(bufX, &sW[0][0][0], sB[0], lane);
  mlp_layer<2, true >(bufX, &sW[1][0][0], sB[1], lane);
  mlp_layer<1, false>(bufX, &sW[2][0][0], sB[2], lane); // shs: 16 outputs
  mlp_layer<2, true >(bufW, &sW[3][0][0], sB[3], lane);
  mlp_layer<2, true >(bufW, &sW[4][0][0], sB[4], lane);
  mlp_layer<2, false>(bufW, &sW[5][0][0], sB[5], lane); // ws: 30 outputs (cols 30,31 junk, unread)

  // ---- phase 3: SH deg-3 eval + per-sample matrix chain + sigmoid ----
  const float dx = ds[3 * sc + 0], dy = ds[3 * sc + 1], dz = ds[3 * sc + 2];
  const float inv = 1.0f / sqrtf(dx * dx + dy * dy + dz * dz);
  const float X = dx * inv, Y = dy * inv, Z = dz * inv;
  const float xx = X * X, yy = Y * Y, zz = Z * Z;
  const float xy = X * Y, yz = Y * Z, xz = X * Z;

  float sh[16];
#pragma unroll
  for (int i = 0; i < 16; ++i) sh[i] = fX[i];
  float wsv[30];
#pragma unroll
  for (int i = 0; i < 30; ++i) wsv[i] = fW[i];

  float r = 0.28209479177387814f * sh[0];
  r = r - 0.4886025119029199f * Y * sh[1] + 0.4886025119029199f * Z * sh[2]
        - 0.4886025119029199f * X * sh[3];
  r = r + 1.0925484305920792f * xy * sh[4]
        - 1.0925484305920792f * yz * sh[5]
        + 0.31539156525252005f * (2.f * zz - xx - yy) * sh[6]
        - 1.0925484305920792f * xz * sh[7]
        + 0.5462742152960396f * (xx - yy) * sh[8];
  r = r - 0.5900435899266435f * Y * (3.f * xx - yy) * sh[9]
        + 2.890611442640554f * xy * Z * sh[10]
        - 0.4570457994644658f * Y * (4.f * zz - xx - yy) * sh[11]
        + 0.3731763325901154f * Z * (2.f * zz - 3.f * xx - 3.f * yy) * sh[12]
        - 0.4570457994644658f * X * (4.f * zz - xx - yy) * sh[13]
        + 1.445305721320277f * Z * (xx - yy) * sh[14]
        - 0.5900435899266435f * X * (xx - yy) * sh[15];

  float c1v[3], c2v[3];
#pragma unroll
  for (int i = 0; i < 3; ++i) c1v[i] = fmaxf(r * wsv[i] + wsv[3 + i], 0.f);
#pragma unroll
  for (int i = 0; i < 3; ++i)
    c2v[i] = fmaxf(c1v[0] * wsv[6 + i] + c1v[1] * wsv[9 + i] + c1v[2] * wsv[12 + i] + wsv[15 + i], 0.f);
  if (s < n) {
#pragma unroll
    for (int i = 0; i < 3; ++i) {
      const float t = c2v[0] * wsv[18 + i] + c2v[1] * wsv[21 + i] + c2v[2] * wsv[24 + i] + wsv[27 + i];
      out[3 * s + i] = 1.0f / (1.0f + expf(-t));
    }
  }
}

extern "C" void kernel_launch(void* const* d_in, const int* in_sizes, int n_in,
                              void* d_out, int out_size, void* d_ws, size_t ws_size,
                              hipStream_t stream) {
  (void)n_in; (void)d_ws; (void)ws_size; (void)out_size;
  const int n = in_sizes[0] / 3; // xs is (n,3)
  dim3 block(BLOCK);
  dim3 grid((n + BLOCK - 1) / BLOCK);
  Hash1SH_fused_kernel<<<grid, block, 0, stream>>>(
      (const float*)d_in[0], (const float*)d_in[1],
      (const float*)d_in[2], (const float*)d_in[3],
      (const float*)d_in[4], (const float*)d_in[5],
      (const float*)d_in[6], (const float*)d_in[7],
      (const float*)d_in[8], (const float*)d_in[9],
      (const float*)d_in[10], (const float*)d_in[11],
      (const float*)d_in[12], (const float*)d_in[13],
      (const float*)d_in[14], (const float*)d_in[15],
      (float*)d_out, n);
}